// MedianPool2d_3599182594367
// MI455X (gfx1250) — compile-verified
//
#include <hip/hip_runtime.h>

// 3x3 median filter, reflect padding, stride 1 -> output shape == input shape.
// Input: 32 x 3 x 512 x 512 fp32  (96 independent 512x512 planes)
// Memory-bound: ~201 MB total traffic -> ~8.6us floor @ 23.3 TB/s.
// CDNA5 path: async global->LDS staging (ASYNCcnt) + wave32 min/max network.

#define IMG 512
#define TILE_W 128
#define TILE_H 16
#define HALO_W (TILE_W + 2)           // 130
#define HALO_H (TILE_H + 2)           // 18
#define LDS_STRIDE (HALO_W + 2)       // 132 floats per LDS row
#define NTHREADS 256
#define ROWS_PER_THREAD 8             // TILE_H / (NTHREADS / TILE_W)

typedef __attribute__((address_space(1))) int* gptr_t;
typedef __attribute__((address_space(3))) int* lptr_t;

__device__ __forceinline__ void sort3(float a, float b, float c,
                                      float& mn, float& md, float& mx) {
  float t1 = fminf(a, b);
  float t2 = fmaxf(a, b);
  mn = fminf(t1, c);
  mx = fmaxf(t2, c);
  md = fmaxf(t1, fminf(t2, c));
}

__device__ __forceinline__ float med3(float a, float b, float c) {
  float t1 = fminf(a, b);
  float t2 = fmaxf(a, b);
  return fmaxf(t1, fminf(t2, c));
}

__global__ __launch_bounds__(NTHREADS) void median3x3_kernel(
    const float* __restrict__ in, float* __restrict__ out) {
  __shared__ float tile[HALO_H * LDS_STRIDE];

  const int plane = blockIdx.z;
  const int tx0 = blockIdx.x * TILE_W;
  const int ty0 = blockIdx.y * TILE_H;
  const float* __restrict__ src = in + (size_t)plane * (IMG * IMG);
  float* __restrict__ dst = out + (size_t)plane * (IMG * IMG);
  const int tid = threadIdx.x;

  // ---- Stage haloed tile (130 x 18) into LDS via async data mover ----
  for (int e = tid; e < HALO_H * HALO_W; e += NTHREADS) {
    int r = e / HALO_W;
    int c = e - r * HALO_W;
    int gy = ty0 + r - 1;
    int gx = tx0 + c - 1;
    // reflect padding (pad=1): -1 -> 1, 512 -> 510
    gy = (gy < 0) ? -gy : ((gy > IMG - 1) ? 2 * (IMG - 1) - gy : gy);
    gx = (gx < 0) ? -gx : ((gx > IMG - 1) ? 2 * (IMG - 1) - gx : gx);
    const float* gp = src + gy * IMG + gx;
    float* lp = &tile[r * LDS_STRIDE + c];
#if __has_builtin(__builtin_amdgcn_global_load_async_to_lds_b32)
    // global addr is AS(1); LDS offset is the low 32 bits of the generic
    // LDS address.
    __builtin_amdgcn_global_load_async_to_lds_b32(
        (gptr_t)(unsigned long long)(const void*)gp,
        (lptr_t)(unsigned int)(unsigned long long)(void*)lp,
        /*offset=*/0, /*cpol=*/0);
#else
    *lp = *gp;
#endif
  }

#if __has_builtin(__builtin_amdgcn_s_wait_asynccnt)
  __builtin_amdgcn_s_wait_asynccnt(0);
#else
  asm volatile("s_wait_asynccnt 0" ::: "memory");
#endif
  __syncthreads();

  // ---- Compute: one column x 8 output rows per thread, rolling triples ----
  const int x = tid & (TILE_W - 1);
  const int ybase = (tid >> 7) * ROWS_PER_THREAD;

  float mn0, md0, mx0, mn1, md1, mx1, mn2, md2, mx2;
  {
    const float* rp = &tile[ybase * LDS_STRIDE + x];
    sort3(rp[0], rp[1], rp[2], mn0, md0, mx0);
    rp += LDS_STRIDE;
    sort3(rp[0], rp[1], rp[2], mn1, md1, mx1);
  }
#pragma unroll
  for (int y = 0; y < ROWS_PER_THREAD; ++y) {
    const float* rp = &tile[(ybase + y + 2) * LDS_STRIDE + x];
    sort3(rp[0], rp[1], rp[2], mn2, md2, mx2);
    // median of 9 = med3( max of mins, med of meds, min of maxes )
    float lo = fmaxf(fmaxf(mn0, mn1), mn2);
    float hi = fminf(fminf(mx0, mx1), mx2);
    float mi = med3(md0, md1, md2);
    dst[(ty0 + ybase + y) * IMG + tx0 + x] = med3(lo, mi, hi);
    mn0 = mn1; md0 = md1; mx0 = mx1;
    mn1 = mn2; md1 = md2; mx1 = mx2;
  }
}

extern "C" void kernel_launch(void* const* d_in, const int* in_sizes, int n_in,
                              void* d_out, int out_size, void* d_ws, size_t ws_size,
                              hipStream_t stream) {
  (void)in_sizes; (void)n_in; (void)out_size; (void)d_ws; (void)ws_size;
  const float* x = (const float*)d_in[0];
  float* out = (float*)d_out;
  dim3 grid(IMG / TILE_W, IMG / TILE_H, 32 * 3);
  median3x3_kernel<<<grid, NTHREADS, 0, stream>>>(x, out);
}